// YoloLoss_1297080123428
// MI455X (gfx1250) — compile-verified
//
#include <hip/hip_runtime.h>
#include <hip/hip_bf16.h>
#include <math.h>

// ---------------------------------------------------------------------------
// YOLO loss on MI455X (gfx1250, wave32).
// Bandwidth-bound streaming reduction (~177 MB -> ~7.6 us roofline @23.3TB/s).
// Cross-lane reduction done with V_WMMA_F32_16X16X4_F32 (exact fp32):
//   A = per-lane partials (lane l owns row l%16), B = ones(4x16)
//   => D[m][*] = partial(lane m) + partial(lane m+16); sum 8 D regs + 1
//   shfl_xor(16) gives the wave total in every lane.
// ---------------------------------------------------------------------------

typedef __attribute__((ext_vector_type(2))) float v2f;
typedef __attribute__((ext_vector_type(8))) float v8f;

#define SDIM 14
#define PRED_C 30
#define CLS_C 20

__device__ __forceinline__ float iou_vs(float x, float y, float w, float h,
                                        float t_x1, float t_y1, float t_x2,
                                        float t_y2, float t_area, float invS) {
  float p_x1 = x * invS - 0.5f * w, p_y1 = y * invS - 0.5f * h;
  float p_x2 = x * invS + 0.5f * w, p_y2 = y * invS + 0.5f * h;
  float ltx = fmaxf(p_x1, t_x1), lty = fmaxf(p_y1, t_y1);
  float rbx = fminf(p_x2, t_x2), rby = fminf(p_y2, t_y2);
  float iw = fmaxf(rbx - ltx, 0.f), ih = fmaxf(rby - lty, 0.f);
  float inter = iw * ih;
  float p_area = (p_x2 - p_x1) * (p_y2 - p_y1);
  return inter / (p_area + t_area - inter);
}

__global__ void yolo_zero_ws(float* __restrict__ ws) {
  if (threadIdx.x < 8) ws[threadIdx.x] = 0.f;
}

// acc channels: 0=cls_sum 1=noobj_sum 2=reg_sum 3=conf_sum 4=n_obj
__global__ __launch_bounds__(256) void yolo_main_kernel(
    const float* __restrict__ pred, const float* __restrict__ tbox,
    const float* __restrict__ tcls, const int* __restrict__ mask,
    float* __restrict__ ws, int total) {
  const float invS = 1.0f / (float)SDIM;
  float acc[5] = {0.f, 0.f, 0.f, 0.f, 0.f};

  int tid = blockIdx.x * blockDim.x + threadIdx.x;
  int stride = gridDim.x * blockDim.x;

  for (int i = tid; i < total; i += stride) {
    // pred cell: 30 floats, 8-byte aligned -> 15x b64 loads
    const float2* pp = reinterpret_cast<const float2*>(pred + (size_t)i * PRED_C);
    float p[PRED_C];
#pragma unroll
    for (int j = 0; j < PRED_C / 2; ++j) {
      float2 v = pp[j];
      p[2 * j] = v.x;
      p[2 * j + 1] = v.y;
    }
    float4 tb = reinterpret_cast<const float4*>(tbox)[i];
    // cls cell: 20 floats = 80B, always 16B aligned -> 5x b128 loads
    const float4* tcp = reinterpret_cast<const float4*>(tcls + (size_t)i * CLS_C);

    float obj = (mask[i] != 0) ? 1.f : 0.f;
    float noobj = 1.f - obj;

    // no-obj term: sum of squares of both full 5-vec boxes
    float ss = 0.f;
#pragma unroll
    for (int j = 0; j < 10; ++j) ss += p[j] * p[j];
    acc[1] += noobj * ss;

    // class term
    float cs = 0.f;
#pragma unroll
    for (int q = 0; q < 5; ++q) {
      float4 t = tcp[q];
      float d0 = p[10 + 4 * q + 0] - t.x;
      float d1 = p[10 + 4 * q + 1] - t.y;
      float d2 = p[10 + 4 * q + 2] - t.z;
      float d3 = p[10 + 4 * q + 3] - t.w;
      cs += d0 * d0 + d1 * d1 + d2 * d2 + d3 * d3;
    }
    acc[0] += obj * cs;

    // target corners (xywh2xyxy)
    float tx = tb.x, ty = tb.y, tw = tb.z, th = tb.w;
    float t_x1 = tx * invS - 0.5f * tw, t_y1 = ty * invS - 0.5f * th;
    float t_x2 = tx * invS + 0.5f * tw, t_y2 = ty * invS + 0.5f * th;
    float t_area = (t_x2 - t_x1) * (t_y2 - t_y1);

    float iou1 = iou_vs(p[0], p[1], p[2], p[3], t_x1, t_y1, t_x2, t_y2, t_area, invS);
    float iou2 = iou_vs(p[5], p[6], p[7], p[8], t_x1, t_y1, t_x2, t_y2, t_area, invS);
    bool take1 = iou1 >= iou2;
    float bx = take1 ? p[0] : p[5];
    float by = take1 ? p[1] : p[6];
    float bw = take1 ? p[2] : p[7];
    float bh = take1 ? p[3] : p[8];
    float bc = take1 ? p[4] : p[9];
    float biou = take1 ? iou1 : iou2;

    float dsw = sqrtf(bw) - sqrtf(tw);
    float dsh = sqrtf(bh) - sqrtf(th);
    float reg = (bx - tx) * (bx - tx) + (by - ty) * (by - ty) + dsw * dsw + dsh * dsh;
    acc[2] += obj * reg;
    float dc = bc - biou;
    acc[3] += obj * dc * dc;
    acc[4] += obj;
  }

  // -------- block reduction: WMMA cross-lane sum, then LDS, then global ----
  __shared__ float lred[5];
  if (threadIdx.x < 5) lred[threadIdx.x] = 0.f;
  __syncthreads();

  unsigned lane = threadIdx.x & 31u;
  v2f bones;
  bones.x = 1.f;
  bones.y = 1.f;

#pragma unroll
  for (int c = 0; c < 5; ++c) {
    v2f a;
    a.x = acc[c];  // A[l%16][k(lane-half)] = partial; other K slot = 0
    a.y = 0.f;
    v8f z = {};
    // D[m][n] = sum_k A[m][k] = acc(lane m) + acc(lane m+16)
    v8f d = __builtin_amdgcn_wmma_f32_16x16x4_f32(
        /*neg_a=*/false, a, /*neg_b=*/false, bones,
        /*c_mod=*/(short)0, z, /*reuse_a=*/false, /*reuse_b=*/false);
    float s = d[0] + d[1] + d[2] + d[3] + d[4] + d[5] + d[6] + d[7];
    s += __shfl_xor(s, 16);  // rows 0-7 half + rows 8-15 half -> wave total
    if (lane == 0) atomicAdd(&lred[c], s);
  }
  __syncthreads();
  if (threadIdx.x < 5) atomicAdd(&ws[threadIdx.x], lred[threadIdx.x]);
}

__global__ void yolo_finalize_kernel(const float* __restrict__ ws,
                                     float* __restrict__ out,
                                     float total_cells, float n_batch) {
  if (threadIdx.x == 0 && blockIdx.x == 0) {
    float cls_sum = ws[0];
    float noobj_sum = ws[1];
    float reg_sum = ws[2];
    float conf_sum = ws[3];
    float n_obj = ws[4];
    float n_noobj = total_cells - n_obj;

    float cls_loss = cls_sum / n_batch;
    float no_obj_loss = 0.5f * noobj_sum / n_noobj;
    float reg_loss = reg_sum * 5.0f / n_obj;
    float conf_loss = conf_sum / n_obj;

    out[0] = reg_loss + conf_loss + no_obj_loss + cls_loss;
    out[1] = reg_loss;
    out[2] = conf_loss;
    out[3] = no_obj_loss;
    out[4] = cls_loss;
  }
}

extern "C" void kernel_launch(void* const* d_in, const int* in_sizes, int n_in,
                              void* d_out, int out_size, void* d_ws,
                              size_t ws_size, hipStream_t stream) {
  const float* pred = (const float*)d_in[0];  // [N,S,S,30]
  const float* tbox = (const float*)d_in[1];  // [N,S,S,4]
  const float* tcls = (const float*)d_in[2];  // [N,S,S,20]
  const int* mask = (const int*)d_in[3];      // [N,S,S] bool->int
  float* out = (float*)d_out;                 // 5 scalars
  float* ws = (float*)d_ws;                   // 8 floats used

  int total = in_sizes[3];                       // N*S*S cells
  int n_batch = in_sizes[0] / (SDIM * SDIM * PRED_C);

  yolo_zero_ws<<<1, 32, 0, stream>>>(ws);

  int blocks = (total + 255) / 256;  // 3136 for the reference shape
  if (blocks > 8192) blocks = 8192;
  yolo_main_kernel<<<blocks, 256, 0, stream>>>(pred, tbox, tcls, mask, ws, total);

  yolo_finalize_kernel<<<1, 32, 0, stream>>>(ws, out, (float)total,
                                             (float)n_batch);
}